// LogNCDE_77214922047707
// MI455X (gfx1250) — compile-verified
//
#include <hip/hip_runtime.h>

// Log-NCDE for MI455X (gfx1250): one workgroup per batch element, in-kernel
// 128-step scan. Jacobian-chain GEMMs run on V_WMMA_F32_16X16X4_F32 (fp32
// matrix engine -> no precision loss across the sequential integration).
// Wv0/Wv1 cached in LDS (bank-conflict-padded), Wv2 streamed from L2 with
// global_prefetch overlap against the scalar MLP phases.

#define DD   8
#define SS   64
#define HH   128
#define OUTD 8
#define TT   2049
#define BB   32
#define WINW 16
#define NWIN 128
#define PP   28

#define W1LD 130     // padded Wv1 LDS row stride (dwords): bank = (2m+k)%64, conflict-free
#define STLD 66      // padded staging row stride (dwords)

typedef __attribute__((ext_vector_type(2))) float v2f;
typedef __attribute__((ext_vector_type(8))) float v8f;

__device__ __forceinline__ float sp_f(float x) {           // softplus (JAX-style)
    return fmaxf(x, 0.0f) + log1pf(expf(-fabsf(x)));
}
__device__ __forceinline__ float sig_f(float x) {          // d softplus / dx
    return 1.0f / (1.0f + expf(-x));
}

// fp32 WMMA, 16x16 tile, K=4 (codegen-confirmed in round 1).
__device__ __forceinline__ v8f wmma4(v2f a, v2f b, v8f c) {
    return __builtin_amdgcn_wmma_f32_16x16x4_f32(
        /*neg_a=*/false, a, /*neg_b=*/false, b,
        /*c_mod=*/(short)0, c, /*reuse_a=*/false, /*reuse_b=*/false);
}

__global__ __launch_bounds__(256) void logncde_kernel(
    const float* __restrict__ x,
    const float* __restrict__ Wi0, const float* __restrict__ bi0,
    const float* __restrict__ Wi1, const float* __restrict__ bi1,
    const float* __restrict__ Wi2, const float* __restrict__ bi2,
    const float* __restrict__ Wv0, const float* __restrict__ bv0,
    const float* __restrict__ Wv1, const float* __restrict__ bv1,
    const float* __restrict__ Wv2, const float* __restrict__ bv2,
    const float* __restrict__ Wr,  const float* __restrict__ br,
    float* __restrict__ out)
{
    // ~202 KB LDS total (320 KB available per WGP on CDNA5)
    __shared__ float s1lds[NWIN * DD];                  //  4 KB
    __shared__ float s2lds[NWIN * PP];                  // 14 KB
    __shared__ float hbuf[2][SS];
    __shared__ float z0lds[HH], g0lds[HH];
    __shared__ float z1lds[HH], g1lds[HH];
    __shared__ float Vlds[DD * SS];                     //  2 KB
    __shared__ float gvlds[DD * SS];                    //  2 KB
    __shared__ float dz1dh[HH * SS];                    // 32 KB
    __shared__ __align__(16) float stage[8][16 * STLD]; // 33 KB per-wave dVdh stripe
    __shared__ float JVlds[DD * SS * DD];               // 16 KB  JV[r][e]
    __shared__ __align__(16) float Wv0s[HH * SS];       // 32 KB  LDS-resident Wv0
    __shared__ __align__(16) float Wv1s[HH * W1LD];     // 65 KB  LDS-resident Wv1 (padded)

    const int tid  = threadIdx.x;
    const int lane = tid & 31;
    const int wave = tid >> 5;
    const int b    = blockIdx.x;

    // ------------- Preload Wv0/Wv1 into LDS (one-time) -----------------------
    {
        float4*       d0 = reinterpret_cast<float4*>(Wv0s);
        const float4* s0 = reinterpret_cast<const float4*>(Wv0);
        for (int i = tid; i < (HH * SS) / 4; i += 256) d0[i] = s0[i];
        const float2* s1p = reinterpret_cast<const float2*>(Wv1);
        for (int i = tid; i < HH * (HH / 2); i += 256) {
            const int row = i >> 6, c2 = i & 63;       // 64 float2 per row
            *reinterpret_cast<float2*>(&Wv1s[row * W1LD + c2 * 2]) = s1p[row * 64 + c2];
        }
    }

    // ------------- Phase A: depth-2 log-signatures (1 window / thread) -------
    if (tid < NWIN) {
        const float* xb = x + ((size_t)b * TT + (size_t)tid * WINW) * DD;
        float prev[DD], cum[DD], s1v[DD], M[DD][DD];
        #pragma unroll
        for (int i = 0; i < DD; ++i) {
            prev[i] = xb[i]; cum[i] = 0.f; s1v[i] = 0.f;
            #pragma unroll
            for (int j = 0; j < DD; ++j) M[i][j] = 0.f;
        }
        for (int w = 1; w <= WINW; ++w) {
            float dl[DD];
            #pragma unroll
            for (int i = 0; i < DD; ++i) {
                float c = xb[w * DD + i];
                dl[i] = c - prev[i]; prev[i] = c;
            }
            #pragma unroll
            for (int i = 0; i < DD; ++i)        // exclusive prefix (cum before add)
                #pragma unroll
                for (int j = 0; j < DD; ++j) M[i][j] += cum[i] * dl[j];
            #pragma unroll
            for (int i = 0; i < DD; ++i) { s1v[i] += dl[i]; cum[i] += dl[i]; }
        }
        #pragma unroll
        for (int i = 0; i < DD; ++i) s1lds[tid * DD + i] = s1v[i];
        int p = 0;
        #pragma unroll
        for (int i = 0; i < DD; ++i)
            #pragma unroll
            for (int j = i + 1; j < DD; ++j, ++p)
                s2lds[tid * PP + p] = 0.5f * (M[i][j] - M[j][i]);
    }
    __syncthreads();

    // ------------- Phase B: initial MLP  h0 = Wi2 sp(Wi1 sp(Wi0 x0)) ---------
    if (tid < HH) {
        const float* x0 = x + (size_t)b * TT * DD;
        float acc = bi0[tid];
        #pragma unroll
        for (int i = 0; i < DD; ++i) acc += Wi0[tid * DD + i] * x0[i];
        z0lds[tid] = sp_f(acc);
    }
    __syncthreads();
    if (tid < HH) {
        float acc = bi1[tid];
        for (int k = 0; k < HH; ++k) acc += Wi1[tid * HH + k] * z0lds[k];
        z1lds[tid] = sp_f(acc);
    }
    __syncthreads();
    if (tid < SS) {
        float acc = bi2[tid];
        for (int k = 0; k < HH; ++k) acc += Wi2[tid * HH + k] * z1lds[k];
        hbuf[0][tid] = acc;
    }
    __syncthreads();
    if (tid < OUTD) {                          // readout at n = 0
        float acc = br[tid];
        for (int a = 0; a < SS; ++a) acc += Wr[tid * SS + a] * hbuf[0][a];
        out[((size_t)b * (NWIN + 1)) * OUTD + tid] = acc;
    }
    __syncthreads();

    // ------------- Phase C: the 128-step log-ODE scan ------------------------
    for (int n = 0; n < NWIN; ++n) {
        const float* hcur  = hbuf[n & 1];
        float*       hnext = hbuf[(n + 1) & 1];

        // Warm the Wv2 stream (256 KB, L2-resident) while scalar phases run:
        // emits global_prefetch_b8, overlaps L2->L0 fill with VALU work.
        {
            const char* base = reinterpret_cast<const char*>(Wv2);
            #pragma unroll
            for (int i = 0; i < 8; ++i)
                __builtin_prefetch(base + (size_t)tid * 128 + (size_t)i * 256 * 128, 0, 3);
        }

        // layer 0: pre0 = Wv0 h + bv0 ; z0 = sp, g0 = sigmoid
        if (tid < HH) {
            float acc = bv0[tid];
            for (int c = 0; c < SS; ++c) acc += Wv0s[tid * SS + c] * hcur[c];
            z0lds[tid] = sp_f(acc);
            g0lds[tid] = sig_f(acc);
        }
        __syncthreads();
        // layer 1
        if (tid < HH) {
            float acc = bv1[tid];
            for (int k = 0; k < HH; ++k) acc += Wv1s[tid * W1LD + k] * z0lds[k];
            z1lds[tid] = sp_f(acc);
            g1lds[tid] = sig_f(acc);
        }
        __syncthreads();
        // layer 2: V = tanh(pre2), gv = 1 - tanh^2   (2 rows / thread)
        for (int r = tid; r < DD * SS; r += 256) {
            float acc = bv2[r];
            for (int k = 0; k < HH; ++k) acc += Wv2[r * HH + k] * z1lds[k];
            float t = tanhf(acc);
            Vlds[r] = t; gvlds[r] = 1.0f - t * t;
        }
        // GEMM1 (WMMA): dz1dh = diag(g1) * Wv1 * (diag(g0)*Wv0)   [128x64]
        {
            const int m  = lane & 15;
            const int kh = lane >> 4;
            const int r0 = wave * 16;          // 8 waves x 16 rows = 128
            v8f acc[4] = { {}, {}, {}, {} };
            for (int k0 = 0; k0 < HH; k0 += 4) {
                const int ka = k0 + 2 * kh;
                v2f af;                        // A-frag: lane = M row, K pair by lane-half
                af.x = Wv1s[(r0 + m) * W1LD + ka];
                af.y = Wv1s[(r0 + m) * W1LD + ka + 1];
                #pragma unroll
                for (int nt = 0; nt < 4; ++nt) {
                    v2f bf;                    // B[k][n] = g0[k] * Wv0[k][n]
                    bf.x = g0lds[ka]     * Wv0s[ka * SS + nt * 16 + m];
                    bf.y = g0lds[ka + 1] * Wv0s[(ka + 1) * SS + nt * 16 + m];
                    acc[nt] = wmma4(af, bf, acc[nt]);
                }
            }
            #pragma unroll
            for (int nt = 0; nt < 4; ++nt)
                #pragma unroll
                for (int j = 0; j < 8; ++j) {  // C/D: vgpr j -> M = kh*8+j, lane -> N
                    const int row = r0 + kh * 8 + j;
                    dz1dh[row * SS + nt * 16 + m] = g1lds[row] * acc[nt][j];
                }
        }
        __syncthreads();
        // GEMM2 (WMMA): dVdh stripe = diag(gv) * Wv2 * dz1dh, staged per wave,
        // then GEMM3 (WMMA): JV stripe = stripe(16x64) @ V^T(64x8)
        {
            const int m  = lane & 15;
            const int kh = lane >> 4;
            float* st = stage[wave];
            for (int q = 0; q < 4; ++q) {      // 4 row-tiles per wave -> 512 rows
                const int r0 = (wave * 4 + q) * 16;
                v8f acc[4] = { {}, {}, {}, {} };
                for (int k0 = 0; k0 < HH; k0 += 4) {
                    const int ka = k0 + 2 * kh;
                    v2f af;
                    af.x = Wv2[(r0 + m) * HH + ka];
                    af.y = Wv2[(r0 + m) * HH + ka + 1];
                    #pragma unroll
                    for (int nt = 0; nt < 4; ++nt) {
                        v2f bf;
                        bf.x = dz1dh[ka * SS + nt * 16 + m];
                        bf.y = dz1dh[(ka + 1) * SS + nt * 16 + m];
                        acc[nt] = wmma4(af, bf, acc[nt]);
                    }
                }
                #pragma unroll
                for (int nt = 0; nt < 4; ++nt)
                    #pragma unroll
                    for (int j = 0; j < 8; ++j) {
                        const int mr = kh * 8 + j;
                        st[mr * STLD + nt * 16 + m] = gvlds[r0 + mr] * acc[nt][j];
                    }
                // GEMM3: contract stripe against V^T (N=8 padded to 16).
                // Same-wave LDS store->load is in-order on CDNA5; no barrier.
                v8f acc3 = {};
                for (int k0 = 0; k0 < SS; k0 += 4) {
                    const int ka = k0 + 2 * kh;
                    v2f af; af.x = st[m * STLD + ka]; af.y = st[m * STLD + ka + 1];
                    v2f bf;                    // Vt[k][e] = V[e*64+k], pad e>=8
                    bf.x = (m < DD) ? Vlds[m * SS + ka]     : 0.0f;
                    bf.y = (m < DD) ? Vlds[m * SS + ka + 1] : 0.0f;
                    acc3 = wmma4(af, bf, acc3);
                }
                if (m < DD) {
                    #pragma unroll
                    for (int j = 0; j < 8; ++j) {
                        const int row = r0 + kh * 8 + j;
                        JVlds[row * DD + m] = acc3[j];
                    }
                }
            }
        }
        __syncthreads();
        // log-ODE update: hn = h + a@V + b@[V_i,V_j]
        if (tid < SS) {
            const int a = tid;
            float acc = hcur[a];
            #pragma unroll
            for (int d = 0; d < DD; ++d) acc += s1lds[n * DD + d] * Vlds[d * SS + a];
            int p = 0;
            #pragma unroll
            for (int i = 0; i < DD; ++i)
                #pragma unroll
                for (int j = i + 1; j < DD; ++j, ++p)
                    acc += s2lds[n * PP + p] *
                           (JVlds[(j * SS + a) * DD + i] - JVlds[(i * SS + a) * DD + j]);
            hnext[a] = acc;
        }
        __syncthreads();
        if (tid < OUTD) {                      // evolving readout
            float acc = br[tid];
            for (int a = 0; a < SS; ++a) acc += Wr[tid * SS + a] * hnext[a];
            out[((size_t)b * (NWIN + 1) + (n + 1)) * OUTD + tid] = acc;
        }
        __syncthreads();
    }
}

extern "C" void kernel_launch(void* const* d_in, const int* in_sizes, int n_in,
                              void* d_out, int out_size, void* d_ws, size_t ws_size,
                              hipStream_t stream) {
    // setup_inputs order: ts, x, Wi0,bi0, Wi1,bi1, Wi2,bi2, Wv0,bv0, Wv1,bv1,
    //                     Wv2,bv2, Wr,br   (ts unused by the reference)
    const float* x   = (const float*)d_in[1];
    const float* Wi0 = (const float*)d_in[2];
    const float* bi0 = (const float*)d_in[3];
    const float* Wi1 = (const float*)d_in[4];
    const float* bi1 = (const float*)d_in[5];
    const float* Wi2 = (const float*)d_in[6];
    const float* bi2 = (const float*)d_in[7];
    const float* Wv0 = (const float*)d_in[8];
    const float* bv0 = (const float*)d_in[9];
    const float* Wv1 = (const float*)d_in[10];
    const float* bv1 = (const float*)d_in[11];
    const float* Wv2 = (const float*)d_in[12];
    const float* bv2 = (const float*)d_in[13];
    const float* Wr  = (const float*)d_in[14];
    const float* br  = (const float*)d_in[15];

    logncde_kernel<<<BB, 256, 0, stream>>>(x, Wi0, bi0, Wi1, bi1, Wi2, bi2,
                                           Wv0, bv0, Wv1, bv1, Wv2, bv2,
                                           Wr, br, (float*)d_out);
}